// Model_9792525435596
// MI455X (gfx1250) — compile-verified
//
#include <hip/hip_runtime.h>
#include <hip/hip_bf16.h>

typedef float v2f __attribute__((ext_vector_type(2)));
typedef float v8f __attribute__((ext_vector_type(8)));

namespace {
constexpr int kB  = 2;
constexpr int kH  = 16;
constexpr int kLQ = 2048;
constexpr int kLK = 2048;
constexpr int kD  = 64;
constexpr int kBH = kB * kH;

// LDS strides (dwords): multiples of 4 (b128-aligned rows) with bank spreads:
//  - K reads: addr = n*68 + 4s + 2g -> banks 4n+2g(+1), all 64 distinct.
//  - V reads: 2*80 = 160 ≡ 32 (mod 64) -> half-wave groups in disjoint ranges.
//  - P tile: stride 18 (even -> 8B-aligned b64 A-fragment reads).
constexpr int KSTR = 68;
constexpr int VSTR = 80;
constexpr int PSTR = 18;

constexpr float kLog2e = 1.4426950408889634f;

__device__ __forceinline__ float fast_rcp(float x) {
#if __has_builtin(__builtin_amdgcn_rcpf)
    return __builtin_amdgcn_rcpf(x);
#else
    return 1.0f / x;
#endif
}
__device__ __forceinline__ float fast_exp2(float x) {
#if __has_builtin(__builtin_amdgcn_exp2f)
    return __builtin_amdgcn_exp2f(x);
#else
    return exp2f(x);
#endif
}

// ---- Cross-lane helpers for the 16-lane row groups ----
// xor 1/2/4 stay within 8-lane groups -> DPP8 (VALU, dual-issuable, no addr).
constexpr unsigned dpp8_sel_xor(int m) {
    unsigned v = 0;
    for (int i = 0; i < 8; ++i) v |= (unsigned)((i ^ m) & 7) << (3 * i);
    return v;
}
template <unsigned SEL>
__device__ __forceinline__ float dpp8_xor_f(float x) {
    return __int_as_float(__builtin_amdgcn_mov_dpp8(__float_as_int(x), SEL));
}
// xor 8: ds_swizzle SWAPX8 (imm-encoded, no address VGPR):
// offset[4:0]=and=0x1f, offset[9:5]=or=0, offset[14:10]=xor=8 -> 0x201F.
__device__ __forceinline__ float swz_xor8_f(float x) {
    return __int_as_float(__builtin_amdgcn_ds_swizzle(__float_as_int(x), 0x201F));
}
__device__ __forceinline__ float rowmax16(float x) {
    x = fmaxf(x, dpp8_xor_f<dpp8_sel_xor(1)>(x));
    x = fmaxf(x, dpp8_xor_f<dpp8_sel_xor(2)>(x));
    x = fmaxf(x, dpp8_xor_f<dpp8_sel_xor(4)>(x));
    x = fmaxf(x, swz_xor8_f(x));
    return x;
}
__device__ __forceinline__ float rowsum16(float x) {
    x += dpp8_xor_f<dpp8_sel_xor(1)>(x);
    x += dpp8_xor_f<dpp8_sel_xor(2)>(x);
    x += dpp8_xor_f<dpp8_sel_xor(4)>(x);
    x += swz_xor8_f(x);
    return x;
}
}

__global__ __launch_bounds__(128)
void fattn_f32_wmma(const float* __restrict__ Q,
                    const float* __restrict__ K,
                    const float* __restrict__ V,
                    const float* __restrict__ invs,
                    const float* __restrict__ mask,
                    float* __restrict__ O)
{
    // Double-buffered K/V tiles so the async DMA of tile t+1 overlaps the
    // entire compute of tile t (single barrier per iteration).
    __shared__ __align__(16) float sK[2][16 * KSTR];
    __shared__ __align__(16) float sV[2][16 * VSTR];
    __shared__ __align__(16) float sP[4][16 * PSTR];

    const int bh   = blockIdx.x;
    const int qt   = blockIdx.y;
    const int wave = threadIdx.x >> 5;
    const int lane = threadIdx.x & 31;
    const int n    = lane & 15;   // column (or A-row) index within half-wave
    const int g    = lane >> 4;   // half-wave select (+2 in K dim, +8 in M dim)

    const int q0 = qt * 64 + wave * 16;

    const float* Qb = Q + ((size_t)bh * kLQ + q0) * kD;
    const float* Kb = K + (size_t)bh * kLK * kD;
    const float* Vb = V + (size_t)bh * kLK * kD;
    const float* Mb = mask + ((size_t)bh * kLQ + q0) * (size_t)kLK;
    float*       Ob = O + ((size_t)bh * kLQ + q0) * kD;

    // Per-thread tile-load geometry: 128 threads x 2 chunks = 16 rows x 64 d.
    const int row0 = (int)threadIdx.x >> 4;
    const int col0 = ((int)threadIdx.x & 15) << 2;
    const int row1 = row0 + 8;

    // ---- Q as WMMA A-fragments (16 steps of 16x4); streamed once -> NT loads
    v2f aq[16];
#pragma unroll
    for (int s = 0; s < 16; ++s)
        aq[s] = __builtin_nontemporal_load(
                    (const v2f*)(Qb + n * kD + 4 * s + 2 * g));

    v8f acc[4];
#pragma unroll
    for (int t = 0; t < 4; ++t)
#pragma unroll
        for (int j = 0; j < 8; ++j) acc[t][j] = 0.f;

    // mrow: row-uniform running max (base-2 domain);
    // lrow: PER-LANE partial sum, cross-lane reduced once in the epilogue.
    float mrow[8], lrow[8];
#pragma unroll
    for (int j = 0; j < 8; ++j) { mrow[j] = -3.0e38f; lrow[j] = 0.f; }

    // ---- Prologue: async-load tile 0 into buffer 0 ----
    {
        const unsigned lk0 = (unsigned)(uintptr_t)&sK[0][row0 * KSTR + col0];
        const unsigned lk1 = (unsigned)(uintptr_t)&sK[0][row1 * KSTR + col0];
        const unsigned lv0 = (unsigned)(uintptr_t)&sV[0][row0 * VSTR + col0];
        const unsigned lv1 = (unsigned)(uintptr_t)&sV[0][row1 * VSTR + col0];
        const float* gk0 = Kb + (size_t)row0 * kD + col0;
        const float* gk1 = Kb + (size_t)row1 * kD + col0;
        const float* gv0 = Vb + (size_t)row0 * kD + col0;
        const float* gv1 = Vb + (size_t)row1 * kD + col0;
        asm volatile("global_load_async_to_lds_b128 %0, %1, off"
                     :: "v"(lk0), "v"(gk0) : "memory");
        asm volatile("global_load_async_to_lds_b128 %0, %1, off"
                     :: "v"(lk1), "v"(gk1) : "memory");
        asm volatile("global_load_async_to_lds_b128 %0, %1, off"
                     :: "v"(lv0), "v"(gv0) : "memory");
        asm volatile("global_load_async_to_lds_b128 %0, %1, off"
                     :: "v"(lv1), "v"(gv1) : "memory");
    }

    for (int it = 0; it < kLK / 16; ++it) {
        const int k0  = it * 16;
        const int cur = it & 1;

        // Tile t in LDS (my asyncs done) ...
        asm volatile("s_wait_asynccnt 0" ::: "memory");
        // ... and everyone's asyncs done + everyone finished reading buffer
        // (it+1)&1 during iteration it-1  ->  safe to reuse it below.
        __syncthreads();

        // ---- Kick off async DMA for tile t+1 into the other buffer ----
        if (k0 + 16 < kLK) {
            const int nxt = cur ^ 1;
            const unsigned lk0 = (unsigned)(uintptr_t)&sK[nxt][row0 * KSTR + col0];
            const unsigned lk1 = (unsigned)(uintptr_t)&sK[nxt][row1 * KSTR + col0];
            const unsigned lv0 = (unsigned)(uintptr_t)&sV[nxt][row0 * VSTR + col0];
            const unsigned lv1 = (unsigned)(uintptr_t)&sV[nxt][row1 * VSTR + col0];
            const float* gk0 = Kb + (size_t)(k0 + 16 + row0) * kD + col0;
            const float* gk1 = Kb + (size_t)(k0 + 16 + row1) * kD + col0;
            const float* gv0 = Vb + (size_t)(k0 + 16 + row0) * kD + col0;
            const float* gv1 = Vb + (size_t)(k0 + 16 + row1) * kD + col0;
            asm volatile("global_load_async_to_lds_b128 %0, %1, off"
                         :: "v"(lk0), "v"(gk0) : "memory");
            asm volatile("global_load_async_to_lds_b128 %0, %1, off"
                         :: "v"(lk1), "v"(gk1) : "memory");
            asm volatile("global_load_async_to_lds_b128 %0, %1, off"
                         :: "v"(lv0), "v"(gv0) : "memory");
            asm volatile("global_load_async_to_lds_b128 %0, %1, off"
                         :: "v"(lv1), "v"(gv1) : "memory");
            if (k0 + 32 < kLK) {  // keep tile t+2 inbound in L2
                __builtin_prefetch(gk0 + 16 * kD, 0, 0);
                __builtin_prefetch(gv0 + 16 * kD, 0, 0);
            }
        }

        // Streaming loads issued early; latency hides under the S-WMMAs.
        const float scale = invs[k0 + n];
        float msk[8];
#pragma unroll
        for (int j = 0; j < 8; ++j)
            msk[j] = __builtin_nontemporal_load(
                         Mb + (size_t)(j + 8 * g) * kLK + k0 + n);

        // ---- S = Q @ K^T : 16 x V_WMMA_F32_16X16X4_F32 ----
        v8f c;
#pragma unroll
        for (int j = 0; j < 8; ++j) c[j] = 0.f;
#pragma unroll
        for (int s = 0; s < 16; ++s) {
            const v2f bk = *(const v2f*)(&sK[cur][n * KSTR + 4 * s + 2 * g]);
            c = __builtin_amdgcn_wmma_f32_16x16x4_f32(false, aq[s], false, bk,
                                                      (short)0, c, false, false);
        }

        // Base-2 logits: one rcp per tile + 8 multiplies.
        const float rsc = kLog2e * fast_rcp(scale);

#pragma unroll
        for (int j = 0; j < 8; ++j) {
            const float sc   = c[j] * rsc;
            const float r    = rowmax16(sc);           // DPP8 x3 + 1 ds_swizzle
            const float mnew = fmaxf(mrow[j], r);
            const float alpha = fast_exp2(mrow[j] - mnew);
            const float p     = fast_exp2(sc - mnew);
            lrow[j] = lrow[j] * alpha + p;             // per-lane partial, UNMASKED
            mrow[j] = mnew;
#pragma unroll
            for (int t = 0; t < 4; ++t) acc[t][j] *= alpha;
            sP[wave][(j + 8 * g) * PSTR + n] = p * msk[j];  // mask only for P@V
        }
        // same-wave LDS RAW ordered by DScnt (compiler inserts wait)

        // ---- acc += P @ V : 16 x V_WMMA_F32_16X16X4_F32 ----
#pragma unroll
        for (int s = 0; s < 4; ++s) {
            const v2f ap = *(const v2f*)(&sP[wave][n * PSTR + 4 * s + 2 * g]);
#pragma unroll
            for (int t = 0; t < 4; ++t) {
                v2f bv;
                bv.x = sV[cur][(4 * s + 2 * g    ) * VSTR + 16 * t + n];
                bv.y = sV[cur][(4 * s + 2 * g + 1) * VSTR + 16 * t + n];
                acc[t] = __builtin_amdgcn_wmma_f32_16x16x4_f32(false, ap, false, bv,
                                                               (short)0, acc[t],
                                                               false, false);
            }
        }
    }

    // ---- Epilogue: one cross-lane sum per row, then NT stores ----
#pragma unroll
    for (int j = 0; j < 8; ++j) {
        const float l  = rowsum16(lrow[j]);
        const float rl = fast_rcp(l);
#pragma unroll
        for (int t = 0; t < 4; ++t)
            __builtin_nontemporal_store(acc[t][j] * rl,
                Ob + (size_t)(j + 8 * g) * kD + 16 * t + n);
    }
}

extern "C" void kernel_launch(void* const* d_in, const int* in_sizes, int n_in,
                              void* d_out, int out_size, void* d_ws, size_t ws_size,
                              hipStream_t stream) {
    (void)in_sizes; (void)n_in; (void)d_ws; (void)ws_size; (void)out_size;
    const float* Q = (const float*)d_in[0];
    const float* K = (const float*)d_in[1];
    const float* V = (const float*)d_in[2];
    const float* S = (const float*)d_in[3];  // inv_scale_factor [LK]
    const float* M = (const float*)d_in[4];  // dropout mask [B,H,LQ,LK]
    float* O = (float*)d_out;

    dim3 grid(kBH, kLQ / 64, 1);
    dim3 block(128, 1, 1);
    hipLaunchKernelGGL(fattn_f32_wmma, grid, block, 0, stream, Q, K, V, S, M, O);
}